// CasMVSNet_64613488001208
// MI455X (gfx1250) — compile-verified
//
#include <hip/hip_runtime.h>
#include <math.h>

#define GG 8
#define CC 32
#define CG 4
#define HH 128
#define WW 128
#define DD 48
#define VV 5
#define HW (HH*WW)
#define DHW (DD*HW)

typedef float v2f __attribute__((ext_vector_type(2)));
typedef float v8f __attribute__((ext_vector_type(8)));

// ---------------------------------------------------------------------------
// Kernel 0: proj[i] = (affine_mats[idx[i+1]] @ affine_mats_inv[idx[0]])[:3,:]
// ---------------------------------------------------------------------------
__global__ void proj_kernel(const float* __restrict__ A,
                            const float* __restrict__ Ainv,
                            const int* __restrict__ idx,
                            float* __restrict__ proj) {
    int t = threadIdx.x;
    if (t >= 48) return;
    int i  = t / 12;
    int rc = t % 12;
    int r = rc / 4, c = rc % 4;
    const float* Ai = A    + idx[i + 1] * 16;
    const float* I0 = Ainv + idx[0]     * 16;
    float s = 0.f;
#pragma unroll
    for (int k = 0; k < 4; ++k) s += Ai[r * 4 + k] * I0[k * 4 + c];
    proj[t] = s;
}

// ---------------------------------------------------------------------------
// Kernel 1: homography warp + bilinear gather + group correlation.
// One thread per (d,h,w); 4 bilinear weights/addresses shared across all 32
// channels, channel stride reached via 24-bit immediate offsets (c*64 KB).
// volume[g,d,h,w] = (1/16) * sum_{j<4} acc[4g+j] * ref[4g+j,h,w]
// ---------------------------------------------------------------------------
__global__ __launch_bounds__(256) void warp_volume_kernel(
    const float* __restrict__ feats,
    const float* __restrict__ dvals,
    const float* __restrict__ proj,
    const int*   __restrict__ idx,
    float* __restrict__ volume) {
    int tid = blockIdx.x * blockDim.x + threadIdx.x;
    int w = tid & (WW - 1);
    int h = (tid >> 7) & (HH - 1);
    int d = tid >> 14;
    float depth = dvals[tid];                 // (D,H,W) flat == tid
    float fx = (float)w, fy = (float)h;

    const float* fr = feats + (size_t)idx[0] * (CC * HW) + h * WW + w;
    __builtin_prefetch(fr, 0, 1);             // global_prefetch_b8 (gfx1250)

    float acc[CC];
#pragma unroll
    for (int c = 0; c < CC; ++c) acc[c] = 0.f;

    for (int v = 0; v < VV - 1; ++v) {
        const float* p = proj + v * 12;       // uniform -> scalar loads
        float sx = (p[0] * fx + p[1] * fy + p[2])  * depth + p[3];
        float sy = (p[4] * fx + p[5] * fy + p[6])  * depth + p[7];
        float sz = (p[8] * fx + p[9] * fy + p[10]) * depth + p[11];
        float z  = (fabsf(sz) < 1e-6f) ? 1e-6f : sz;
        float rz = 1.0f / z;
        float gx = sx * rz, gy = sy * rz;

        float x0f = floorf(gx), y0f = floorf(gy);
        float wx1 = gx - x0f, wy1 = gy - y0f;
        float wx0 = 1.f - wx1, wy0 = 1.f - wy1;
        int x0 = (int)x0f, y0 = (int)y0f;
        int x1 = x0 + 1,   y1 = y0 + 1;
        float mx0 = (x0 >= 0 && x0 < WW) ? 1.f : 0.f;
        float mx1 = (x1 >= 0 && x1 < WW) ? 1.f : 0.f;
        float my0 = (y0 >= 0 && y0 < HH) ? 1.f : 0.f;
        float my1 = (y1 >= 0 && y1 < HH) ? 1.f : 0.f;
        int x0c = min(max(x0, 0), WW - 1), x1c = min(max(x1, 0), WW - 1);
        int y0c = min(max(y0, 0), HH - 1), y1c = min(max(y1, 0), HH - 1);
        float w00 = wx0 * wy0 * mx0 * my0;
        float w10 = wx1 * wy0 * mx1 * my0;
        float w01 = wx0 * wy1 * mx0 * my1;
        float w11 = wx1 * wy1 * mx1 * my1;
        int o00 = y0c * WW + x0c, o10 = y0c * WW + x1c;
        int o01 = y1c * WW + x0c, o11 = y1c * WW + x1c;

        const float* fv = feats + (size_t)idx[v + 1] * (CC * HW);
#pragma unroll
        for (int c = 0; c < CC; ++c) {
            int cb = c * HW;                  // byte offset c*65536 < 2^23 -> imm
            acc[c] += w00 * fv[cb + o00] + w10 * fv[cb + o10]
                    + w01 * fv[cb + o01] + w11 * fv[cb + o11];
        }
    }

#pragma unroll
    for (int g = 0; g < GG; ++g) {
        float s = 0.f;
#pragma unroll
        for (int j = 0; j < CG; ++j) {
            int c = g * CG + j;
            s += acc[c] * fr[c * HW];
        }
        volume[((g * DD + d) * HH + h) * WW + w] = s * (1.0f / 16.0f);
    }
}

// ---------------------------------------------------------------------------
// Kernel 2: 3D conv (8ch x 3x3x3 -> 1ch, SAME) as implicit GEMM on
// v_wmma_f32_16x16x4_f32, then wave32 softmax-expectation over D.
//
// K axis padded to 288 = 72 chunks of 4: chunk = (g,kd,kh) (compile-time via
// full unroll), K-row within chunk = kw' in {0,1,2,3} with kw'=3 a zero tap.
// Lane supplies A[m][kw'] for kw' = 2*hiHalf + u -> no integer div/mod at all.
// A layout (16x4 f32, ISA 7.12.2): lane<16 -> M=lane, K=0/1; lane>=16 -> K=2/3.
// C/D: VGPR r -> M=r (lanes 0-15) / M=r+8 (lanes 16-31).
// Zero-padding handled by clamp+multiply-mask so EXEC stays all-1s for WMMA.
// One wave per pixel; block = 4 waves.
// ---------------------------------------------------------------------------
__global__ __launch_bounds__(128) void conv_softmax_kernel(
    const float* __restrict__ vol,
    const float* __restrict__ dvals,
    const float* __restrict__ cw,
    const float* __restrict__ cb,
    float* __restrict__ depth_out) {
    __shared__ float s_w4[288];               // weights padded to 4 per (g,kd,kh)
    __shared__ float s_prob[4][DD];

    int tid = threadIdx.x;
    for (int i = tid; i < 288; i += 128) {
        int chunk = i >> 2, kw = i & 3;
        s_w4[i] = (kw < 3) ? cw[chunk * 3 + kw] : 0.f;
    }
    __syncthreads();

    int wave = tid >> 5;
    int lane = tid & 31;
    int pix  = blockIdx.x * 4 + wave;
    int h = pix >> 7;
    int w = pix & 127;
    int m  = lane & 15;
    int hi = lane >> 4;                       // 0 -> K rows 0/1, 1 -> K rows 2/3
    float bias = cb[0];

    // spatial clamps + masks (registers after unroll; constant indices)
    int   hOff[3] = { max(h - 1, 0) * WW, h * WW, min(h + 1, HH - 1) * WW };
    float hMsk[3] = { h > 0 ? 1.f : 0.f, 1.f, h < HH - 1 ? 1.f : 0.f };
    int   wCl[3]  = { max(w - 1, 0), w, min(w + 1, WW - 1) };
    float wMsk[3] = { w > 0 ? 1.f : 0.f, 1.f, w < WW - 1 ? 1.f : 0.f };

    // per-lane kw' selections: u=0 -> kw' = 2*hi ; u=1 -> kw' = 2*hi+1
    int   wOffA = hi ? wCl[2] : wCl[0];
    float wMskA = hi ? wMsk[2] : wMsk[0];
    int   wOffB = wCl[1];                     // kw'=1 (hi=0); kw'=3 zero tap (hi=1)
    float wMskB = hi ? 0.f : wMsk[1];

    for (int dt = 0; dt < 3; ++dt) {
        int dm = dt * 16 + m;                 // output depth for this lane's M row
        int   dOff[3] = { max(dm - 1, 0) * HW, dm * HW, min(dm + 1, DD - 1) * HW };
        float dMsk[3] = { dm > 0 ? 1.f : 0.f, 1.f, dm < DD - 1 ? 1.f : 0.f };

        v8f c = {0.f, 0.f, 0.f, 0.f, 0.f, 0.f, 0.f, 0.f};
#pragma unroll
        for (int g = 0; g < GG; ++g) {
#pragma unroll
            for (int kd = 0; kd < 3; ++kd) {
#pragma unroll
                for (int kh = 0; kh < 3; ++kh) {
                    int chunk = (g * 3 + kd) * 3 + kh;
                    int base  = g * DHW + dOff[kd] + hOff[kh];
                    float mdh = dMsk[kd] * hMsk[kh];
                    v2f a, b;
                    a.x = vol[base + wOffA] * (mdh * wMskA);
                    a.y = vol[base + wOffB] * (mdh * wMskB);
                    b.x = s_w4[chunk * 4 + hi * 2];       // ds_load_b64 pair
                    b.y = s_w4[chunk * 4 + hi * 2 + 1];
                    c = __builtin_amdgcn_wmma_f32_16x16x4_f32(
                            false, a, false, b, (short)0, c, false, false);
                }
            }
        }
        // all N columns identical; lanes 0 / 16 extract M = 0..7 / 8..15
        if ((lane & 15) == 0) {
#pragma unroll
            for (int r = 0; r < 8; ++r)
                s_prob[wave][dt * 16 + hi * 8 + r] = c[r] + bias;
        }
    }
    __syncthreads();

    // wave32 softmax expectation over 48 depths (lanes cover d and d+32)
    float a0 = s_prob[wave][lane];
    float a1 = (lane < 16) ? s_prob[wave][32 + lane] : -INFINITY;
    float mx = fmaxf(a0, a1);
#pragma unroll
    for (int s = 16; s >= 1; s >>= 1) mx = fmaxf(mx, __shfl_xor(mx, s, 32));
    float dv0 = dvals[(lane * HH + h) * WW + w];
    float dv1 = (lane < 16) ? dvals[((32 + lane) * HH + h) * WW + w] : 0.f;
    float e0 = __expf(a0 - mx);
    float e1 = (lane < 16) ? __expf(a1 - mx) : 0.f;
    float se = e0 + e1;
    float sn = e0 * dv0 + e1 * dv1;
#pragma unroll
    for (int s = 16; s >= 1; s >>= 1) {
        se += __shfl_xor(se, s, 32);
        sn += __shfl_xor(sn, s, 32);
    }
    if (lane == 0) depth_out[pix] = sn / se;
}

// ---------------------------------------------------------------------------
extern "C" void kernel_launch(void* const* d_in, const int* in_sizes, int n_in,
                              void* d_out, int out_size, void* d_ws, size_t ws_size,
                              hipStream_t stream) {
    const float* feats = (const float*)d_in[0];
    const float* am    = (const float*)d_in[1];
    const float* ami   = (const float*)d_in[2];
    const float* dvals = (const float*)d_in[3];
    const float* cw    = (const float*)d_in[4];
    const float* cb    = (const float*)d_in[5];
    const int*   idx   = (const int*)d_in[6];

    float* out    = (float*)d_out;
    float* volume = out;                      // 8*48*128*128
    float* depth  = out + (size_t)GG * DD * HW;
    float* proj   = (float*)d_ws;             // 4 views * 12 floats

    proj_kernel<<<1, 64, 0, stream>>>(am, ami, idx, proj);
    warp_volume_kernel<<<(DD * HW) / 256, 256, 0, stream>>>(feats, dvals, proj,
                                                            idx, volume);
    conv_softmax_kernel<<<HW / 4, 128, 0, stream>>>(volume, dvals, cw, cb, depth);
}